// ChamferDistance_43095701848170
// MI455X (gfx1250) — compile-verified
//
#include <hip/hip_runtime.h>
#include <stdint.h>

typedef __attribute__((ext_vector_type(2))) float v2f;
typedef __attribute__((ext_vector_type(8))) float v8f;

// Order-preserving float<->uint mapping so integer atomic_min == float min.
__device__ __forceinline__ unsigned f2u_ord(float f) {
    unsigned b = __float_as_uint(f);
    return (b & 0x80000000u) ? ~b : (b | 0x80000000u);
}
__device__ __forceinline__ float u2f_ord(unsigned u) {
    unsigned b = (u & 0x80000000u) ? (u & 0x7FFFFFFFu) : ~u;
    return __uint_as_float(b);
}

// ---------------------------------------------------------------- init mins
__global__ void cd_init_kernel(unsigned* __restrict__ mins, int n) {
    int i = blockIdx.x * blockDim.x + threadIdx.x;
    if (i < n) mins[i] = 0xFF800000u;   // f2u_ord(+inf)
}

// ---------------------------------------------------------------- main WMMA
// Each wave: 64 gt rows (4 x 16-row A tiles), scans colsPerSplit gen columns.
__global__ __launch_bounds__(256) void cd_main_kernel(
    const float* __restrict__ gt, const float* __restrict__ gen,
    unsigned* __restrict__ rowmin, unsigned* __restrict__ colmin,
    int colsPerSplit) {
    const int lane = threadIdx.x & 31;
    const int wave = threadIdx.x >> 5;
    const int half = lane >> 4;          // K-half for A/B, M/N-half for C/D
    const int m16  = lane & 15;
    const int rowBase  = (blockIdx.x * 8 + wave) * 64;
    const int colStart = blockIdx.y * colsPerSplit;

    // A tiles (16x4 fp32, K padded 3->4) + gt squared norms per C/D slot.
    v2f   a[4];
    float gs[4][8];
    float rm[4][8];
    #pragma unroll
    for (int rt = 0; rt < 4; ++rt) {
        const int arow = rowBase + rt * 16 + m16;
        const float ax = gt[3 * arow + 0];
        const float ay = gt[3 * arow + 1];
        const float az = gt[3 * arow + 2];
        v2f av;
        if (half) { av.x = az; av.y = 0.0f; }
        else      { av.x = ax; av.y = ay;   }
        a[rt] = av;
        #pragma unroll
        for (int r = 0; r < 8; ++r) {
            const int row = rowBase + rt * 16 + r + 8 * half;
            const float x = gt[3 * row + 0];
            const float y = gt[3 * row + 1];
            const float z = gt[3 * row + 2];
            gs[rt][r] = x * x + y * y + z * z;
            rm[rt][r] = __builtin_inff();
        }
    }

    const int nColTiles = colsPerSplit >> 4;

    // software pipeline: preload first B tile
    float bx = gen[3 * (colStart + m16) + 0];
    float by = gen[3 * (colStart + m16) + 1];
    float bz = gen[3 * (colStart + m16) + 2];

    for (int ct = 0; ct < nColTiles; ++ct) {
        const int col = colStart + ct * 16 + m16;

        // prefetch next tile (wraps to tile 0 on last iter; value unused)
        const int nt   = (ct + 1 < nColTiles) ? (ct + 1) : 0;
        const int ncol = colStart + nt * 16 + m16;
        const float nx = gen[3 * ncol + 0];
        const float ny = gen[3 * ncol + 1];
        const float nz = gen[3 * ncol + 2];

        const float sqg = bx * bx + by * by + bz * bz;
        v2f b;
        if (half) { b.x = bz; b.y = 0.0f; }
        else      { b.x = bx; b.y = by;   }

        float cmrt[4];
        #pragma unroll
        for (int rt = 0; rt < 4; ++rt) {
            v8f c = {0.f, 0.f, 0.f, 0.f, 0.f, 0.f, 0.f, 0.f};
            v8f d = __builtin_amdgcn_wmma_f32_16x16x4_f32(
                false, a[rt], false, b, (short)0, c, false, false);
            // two independent FMAs per element; tree-reduce the col-min
            float t[8];
            #pragma unroll
            for (int r = 0; r < 8; ++r) {
                rm[rt][r] = fminf(rm[rt][r], fmaf(-2.0f, d[r], sqg));
                t[r]      = fmaf(-2.0f, d[r], gs[rt][r]);
            }
            float m01 = fminf(t[0], t[1]);
            float m23 = fminf(t[2], t[3]);
            float m45 = fminf(t[4], t[5]);
            float m67 = fminf(t[6], t[7]);
            cmrt[rt] = fminf(fminf(m01, m23), fminf(m45, m67));
        }
        float cm = fminf(fminf(cmrt[0], cmrt[1]), fminf(cmrt[2], cmrt[3]));

        // combine the two 16-lane halves (rows 0..7+8h vs 8..15+8h per slot)
        cm = fminf(cm, __shfl_xor(cm, 16, 32));
        if (lane < 16) atomicMin(&colmin[col], f2u_ord(cm));

        bx = nx; by = ny; bz = nz;
    }

    // finish row mins: butterfly within each 16-lane half (columns of tile)
    #pragma unroll
    for (int rt = 0; rt < 4; ++rt) {
        #pragma unroll
        for (int r = 0; r < 8; ++r) {
            float v = rm[rt][r];
            v = fminf(v, __shfl_xor(v, 1, 32));
            v = fminf(v, __shfl_xor(v, 2, 32));
            v = fminf(v, __shfl_xor(v, 4, 32));
            v = fminf(v, __shfl_xor(v, 8, 32));
            if ((lane & 15) == 0)
                atomicMin(&rowmin[rowBase + rt * 16 + r + 8 * half],
                          f2u_ord(v));
        }
    }
}

// ---------------------------------------------------------------- finalize
__global__ __launch_bounds__(256) void cd_final_kernel(
    const float* __restrict__ gt, const float* __restrict__ gen,
    const unsigned* __restrict__ rowmin, const unsigned* __restrict__ colmin,
    float* __restrict__ out, int N) {
    __shared__ float sdata[256];
    float acc = 0.0f;
    for (int i = threadIdx.x; i < N; i += 256) {
        float x = gt[3 * i + 0], y = gt[3 * i + 1], z = gt[3 * i + 2];
        float v = u2f_ord(rowmin[i]) + (x * x + y * y + z * z);
        acc += fmaxf(v, 0.0f);
        x = gen[3 * i + 0]; y = gen[3 * i + 1]; z = gen[3 * i + 2];
        v = u2f_ord(colmin[i]) + (x * x + y * y + z * z);
        acc += fmaxf(v, 0.0f);
    }
    sdata[threadIdx.x] = acc;
    __syncthreads();
    for (int s = 128; s > 0; s >>= 1) {
        if (threadIdx.x < s) sdata[threadIdx.x] += sdata[threadIdx.x + s];
        __syncthreads();
    }
    if (threadIdx.x == 0) out[0] = sdata[0] / (float)N;
}

extern "C" void kernel_launch(void* const* d_in, const int* in_sizes, int n_in,
                              void* d_out, int out_size, void* d_ws, size_t ws_size,
                              hipStream_t stream) {
    const float* gt  = (const float*)d_in[0];
    const float* gen = (const float*)d_in[1];
    const int N = in_sizes[0] / 3;                 // 8192
    unsigned* rowmin = (unsigned*)d_ws;
    unsigned* colmin = rowmin + N;

    cd_init_kernel<<<(2 * N + 255) / 256, 256, 0, stream>>>(rowmin, 2 * N);

    const int rowGroups    = N / 64;               // 128 row groups (1 per wave)
    const int colSplits    = 8;
    const int colsPerSplit = N / colSplits;        // 1024 cols per wave
    dim3 grid(rowGroups / 8, colSplits);           // 8 waves per 256-thread block
    cd_main_kernel<<<grid, 256, 0, stream>>>(gt, gen, rowmin, colmin,
                                             colsPerSplit);

    cd_final_kernel<<<1, 256, 0, stream>>>(gt, gen, rowmin, colmin,
                                           (float*)d_out, N);
}